// fofe_tricontext_79001628443164
// MI455X (gfx1250) — compile-verified
//
#include <hip/hip_runtime.h>

// FOFE tri-context = left/right first-order alpha-scans (Lf, Rf) + gather.
// Roofline: output write 330 MB / 23.3 TB/s ~ 14 us dominates; phase 1 is
// 6.6 MB in / 13 MB out. Phase 1: one 2-wave block per (batch, 16-dim tile)
// async-stages its 809x16 f32 strip (51.8 KB) into LDS via
// global_load_async_to_lds_b128, then each wave runs a blocked prefix scan
// with chained V_WMMA_F32_16X16X4_F32 (4 K-slices per 16-row tile), carry
// propagated through the WMMA C-layout with a single branch-free wave32
// ds_bpermute per tile.

#define B_    8
#define L_    809
#define D_    256
#define NCAND 8045
#define LPAD  816           // 51 full tiles; removes all bound guards
#define ALPHA_F 0.9f

typedef __attribute__((ext_vector_type(2))) float v2f;
typedef __attribute__((ext_vector_type(8))) float v8f;

// alpha^k, k = 0..17 (indexed only with unroll-time constants -> folds)
__device__ constexpr float APW[18] = {
  1.0f, 0.9f, 0.81f, 0.729f, 0.6561f, 0.59049f, 0.531441f, 0.4782969f,
  0.43046721f, 0.387420489f, 0.3486784401f, 0.31381059609f,
  0.282429536481f, 0.2541865828329f, 0.22876792454961f,
  0.205891132094649f, 0.1853020188851841f, 0.16677181699666568f };

// ---------------------------------------------------------------------------
// Phase 1: LDS-staged blocked scans via WMMA. Block = 64 threads (2 waves):
// wave 0 -> Lf (lower-tri T, forward carry), wave 1 -> Rf (upper-tri U,
// backward carry). Grid = B_ * (D_/16) = 128 blocks.
// A (16x4 f32): lanes 0-15 M=0..15 hold K={0,1}; lanes 16-31 K={2,3}.
// B (4x16):     lanes hold N=lane%16; VGPR0/1 = K rows {0,1}|{2,3}.
// C/D (16x16):  lane = N; VGPR r -> M=r (lanes<16) / M=8+r (lanes>=16).
// ---------------------------------------------------------------------------
__global__ __launch_bounds__(64) void fofe_scan_wmma(const float* __restrict__ x,
                                                     float* __restrict__ Lf,
                                                     float* __restrict__ Rf) {
  __shared__ float strip[LPAD * 16];        // 52,224 B: one (L,16) dim strip

  const int tid   = threadIdx.x;
  const int lane  = tid & 31;
  const int dir   = tid >> 5;               // wave 0: left, wave 1: right
  const int n     = lane & 15;              // column (dim within tile)
  const int mbase = (lane < 16) ? 0 : 8;    // C/D row base per lane half
  const int koff  = (lane < 16) ? 0 : 2;    // A/B K-slice offset per lane half
  const int m     = n;                      // A-matrix row for this lane

  const int b  = blockIdx.x >> 4;           // D_/16 == 16 dim tiles
  const int d0 = (blockIdx.x & 15) * 16;

  // Zero-fill pad rows 809..812 (64 floats, one per thread); rows 813..815 are not written.
  if (tid < (LPAD - L_) * 16) strip[L_ * 16 + tid] = 0.f;

  // Async-stage the strip: 809 rows x 64 B  =  3236 x b128 chunks.
  const float* gsrc = x + (size_t)b * L_ * D_ + d0;
  for (int idx = tid; idx < L_ * 4; idx += 64) {
    const int row = idx >> 2, q = idx & 3;
    const unsigned long long ga =
        (unsigned long long)(uintptr_t)(gsrc + (size_t)row * D_ + q * 4);
    const unsigned lo = (unsigned)(uintptr_t)(&strip[row * 16 + q * 4]);
    asm volatile("global_load_async_to_lds_b128 %0, %1, off"
                 :: "v"(lo), "v"(ga) : "memory");
  }
  asm volatile("s_wait_asynccnt 0" ::: "memory");
  __syncthreads();                          // both waves' data visible

  // Constant triangular alpha-power matrix as 4 K=4 slices of A.
  const float pm = __powf(ALPHA_F, (float)m);   // alpha^m (setup only)
  v2f aT[4];
#pragma unroll
  for (int k = 0; k < 4; ++k) {
    const int   c0  = 4 * k + koff, c1 = c0 + 1;
    const float pc0 = (lane < 16) ? APW[4 * k]     : APW[4 * k + 2];
    const float pc1 = (lane < 16) ? APW[4 * k + 1] : APW[4 * k + 3];
    if (dir == 0) {                         // T[m][c] = a^(m-c), c<=m
      aT[k].x = (m >= c0) ? pm / pc0 : 0.f;
      aT[k].y = (m >= c1) ? pm / pc1 : 0.f;
    } else {                                // U[m][c] = a^(c-m), c>=m
      aT[k].x = (c0 >= m) ? pc0 / pm : 0.f;
      aT[k].y = (c1 >= m) ? pc1 / pm : 0.f;
    }
  }

  float* Ob = (dir == 0 ? Lf : Rf) + (size_t)b * LPAD * D_ + d0;

  // Branch-free carry-shuffle source lane: row 15 lives in lanes 16..31
  // (left scan), row 0 in lanes 0..15 (right scan).
  const int srcl = (dir == 0) ? (16 + n) : n;

  const int ntiles = LPAD / 16;             // 51
  float carry = 0.f;
  for (int t = 0; t < ntiles; ++t) {
    const int base = (dir == 0) ? t * 16 : (ntiles - 1 - t) * 16;

    v2f bX[4];                              // B operand from LDS (guard-free)
#pragma unroll
    for (int k = 0; k < 4; ++k) {
      const int g0 = base + 4 * k + koff;
      bX[k].x = strip[g0 * 16 + n];
      bX[k].y = strip[(g0 + 1) * 16 + n];
    }

    v8f c = {};
#pragma unroll
    for (int k = 0; k < 4; ++k)
      c = __builtin_amdgcn_wmma_f32_16x16x4_f32(false, aT[k], false, bX[k],
                                                (short)0, c, false, false);

    // Cross-tile carry injection + guard-free stores (padded workspace).
#pragma unroll
    for (int r = 0; r < 8; ++r) {
      const float cfL = (lane < 16) ? APW[r + 1]  : APW[r + 9];   // a^(m+1)
      const float cfR = (lane < 16) ? APW[16 - r] : APW[8 - r];   // a^(16-m)
      const float v   = c[r] + ((dir == 0) ? cfL : cfR) * carry;
      c[r] = v;
      Ob[(size_t)(base + mbase + r) * D_ + n] = v;
    }

    // Single ds_bpermute, no exec divergence: select value + lane by cndmask.
    const float cv = (dir == 0) ? c[7] : c[0];
    carry = __shfl(cv, srcl, 32);
  }
}

// ---------------------------------------------------------------------------
// Phase 2: per-candidate gather + streaming non-temporal output writes.
// grid (NCAND, B_), 256 threads = one dim each; 5 coalesced 1 KB NT stores.
// ---------------------------------------------------------------------------
__global__ __launch_bounds__(256) void fofe_gather(const float* __restrict__ Lf,
                                                   const float* __restrict__ Rf,
                                                   float* __restrict__ out) {
  const int nc = blockIdx.x;
  const int b  = blockIdx.y;
  const int d  = threadIdx.x;

  // Decode candidate index -> (s, e). Starts 0..799 have 10 spans each.
  int s, e;
  if (nc < 8000) {
    s = nc / 10;
    e = s + (nc % 10);
  } else {
    int r = nc - 8000;          // tail: start 800 has 9 spans, 801 has 8, ...
    s = 800;
    int len = 9;
    while (r >= len) { r -= len; ++s; --len; }
    e = s + r;
  }

  const float* Lfb = Lf + (size_t)b * LPAD * D_;
  const float* Rfb = Rf + (size_t)b * LPAD * D_;

  const float o1 = (s >= 1)     ? Lfb[(size_t)(s - 1) * D_ + d] : 0.f; // buf1
  const float o2 =                Lfb[(size_t)e * D_ + d];             // buf2
  const float o3 = (e + 1 < L_) ? Rfb[(size_t)(e + 1) * D_ + d] : 0.f; // buf3
  const float o4 =                Rfb[(size_t)s * D_ + d];             // buf4

  float apow = 1.f;                       // alpha^(e-s+1), span <= 9
  for (int i = 0; i <= e - s; ++i) apow *= ALPHA_F;
  const float o0 = o2 - apow * o1;                                     // buf0

  float* op = out + (((size_t)b * NCAND + nc) * 5) * D_ + d;
  __builtin_nontemporal_store(o0, op);
  __builtin_nontemporal_store(o1, op + D_);
  __builtin_nontemporal_store(o2, op + 2 * D_);
  __builtin_nontemporal_store(o3, op + 3 * D_);
  __builtin_nontemporal_store(o4, op + 4 * D_);
}

extern "C" void kernel_launch(void* const* d_in, const int* in_sizes, int n_in,
                              void* d_out, int out_size, void* d_ws, size_t ws_size,
                              hipStream_t stream) {
  const float* x = (const float*)d_in[0];       // (B, L, D) f32; x_mask unused
  float* Lf  = (float*)d_ws;                    // B*LPAD*D floats = 6.69 MB
  float* Rf  = Lf + (size_t)B_ * LPAD * D_;     // + 6.69 MB (ws ~13.4 MB)
  float* out = (float*)d_out;                   // (B, NCAND, 5*D) f32

  // Phase 1: 8 batches * 16 dim-tiles = 128 blocks, 2 waves each (L & R scan).
  fofe_scan_wmma<<<dim3(B_ * 16), dim3(64), 0, stream>>>(x, Lf, Rf);
  // Phase 2: output streaming.
  fofe_gather<<<dim3(NCAND, B_), dim3(256), 0, stream>>>(Lf, Rf, out);
}